// MultiSemMed_69483980915300
// MI455X (gfx1250) — compile-verified
//
#include <hip/hip_runtime.h>
#include <math.h>

typedef float v2f __attribute__((ext_vector_type(2)));
typedef float v8f __attribute__((ext_vector_type(8)));
typedef int   v4i_t __attribute__((vector_size(16)));           // matches builtin's pointee type
typedef __attribute__((address_space(1))) v4i_t* gv4i_p;        // global int4*
typedef __attribute__((address_space(3))) v4i_t* lv4i_p;        // LDS int4*

#if defined(__gfx1250__) && __has_builtin(__builtin_amdgcn_global_load_async_to_lds_b128) && \
    __has_builtin(__builtin_amdgcn_s_wait_asynccnt)
#define USE_ASYNC_LDS 1
#else
#define USE_ASYNC_LDS 0
#endif

// ---------------------------------------------------------------------------
// Generic fp32 WMMA GEMM:  C(MxN) = A(MxK) @ B(KxN), all row-major.
// Block = 128 threads (4 waves). Each block: 16 rows of C x 64 cols.
// Wave w handles the 16x16 tile at cols blockIdx.y*64 + w*16.
// K is processed in chunks of 64; the A slab (16x64 f32) is staged into LDS
// via GLOBAL_LOAD_ASYNC_TO_LDS_B128 (ASYNCcnt) when available, else via VGPRs.
// Uses V_WMMA_F32_16X16X4_F32 (ISA 7.12): A 16x4, B 4x16, C 16x16.
// Fragment layouts per ISA 7.12.2:
//   A: lane l (m=l&15): VGPR0=A[m][k0+kk], VGPR1=A[m][k0+kk+1], kk=(l>>4)*2
//   B: lane l (n=l&15): VGPR0=B[k0+kk][n], VGPR1=B[k0+kk+1][n]
//   C: VGPR r -> row (r + (l>>4)*8), col (l&15)
// ---------------------------------------------------------------------------
#define A_STRIDE 68  // padded LDS row stride (68*4 = 272B = 16B-aligned)

__global__ __launch_bounds__(128)
void gemm_f32_wmma(const float* __restrict__ A, const float* __restrict__ B,
                   float* __restrict__ C, int M, int N, int K) {
  __shared__ float As[16 * A_STRIDE];
  const int lane = threadIdx.x & 31;
  const int wave = threadIdx.x >> 5;
  const int row0 = blockIdx.x * 16;
  const int col0 = blockIdx.y * 64 + wave * 16;
  const int m    = lane & 15;
  const int kk   = (lane >> 4) * 2;    // 0 or 2: which K-pair this half-wave owns
  const int n    = col0 + (lane & 15);

#if USE_ASYNC_LDS
  // 16x64 slab = 256 x 16B chunks; 2 b128 DMA transfers per thread.
  const int ch0 = threadIdx.x;         // chunk ids 0..127
  const int ch1 = threadIdx.x + 128;   // chunk ids 128..255
  const int r0 = ch0 >> 4, c0 = (ch0 & 15) << 2;
  const int r1 = ch1 >> 4, c1 = (ch1 & 15) << 2;
#endif

  v8f acc = {};
  for (int k0 = 0; k0 < K; k0 += 64) {
#if USE_ASYNC_LDS
    __builtin_amdgcn_global_load_async_to_lds_b128(
        (gv4i_p)(A + (size_t)(row0 + r0) * K + k0 + c0),
        (lv4i_p)&As[r0 * A_STRIDE + c0], 0, 0);
    __builtin_amdgcn_global_load_async_to_lds_b128(
        (gv4i_p)(A + (size_t)(row0 + r1) * K + k0 + c1),
        (lv4i_p)&As[r1 * A_STRIDE + c1], 0, 0);
    __builtin_amdgcn_s_wait_asynccnt(0);
#else
    for (int i = threadIdx.x; i < 16 * 64; i += 128) {
      int r = i >> 6, c = i & 63;
      As[r * A_STRIDE + c] = A[(size_t)(row0 + r) * K + k0 + c];
    }
#endif
    if (k0 + 64 < K)  // hint next A slab toward the caches (global_prefetch_b8)
      __builtin_prefetch(&A[(size_t)(row0 + m) * K + k0 + 64 + kk * 8], 0, 0);
    __syncthreads();
#pragma unroll
    for (int ks = 0; ks < 64; ks += 4) {
      v2f a, b;
      a.x = As[m * A_STRIDE + ks + kk];
      a.y = As[m * A_STRIDE + ks + kk + 1];
      const float* bp = B + (size_t)(k0 + ks + kk) * N + n;
      b.x = bp[0];
      b.y = bp[N];
      acc = __builtin_amdgcn_wmma_f32_16x16x4_f32(false, a, false, b,
                                                  (short)0, acc, false, false);
    }
    __syncthreads();
  }
  const int rowoff = (lane >> 4) * 8;
#pragma unroll
  for (int r = 0; r < 8; ++r)
    C[(size_t)(row0 + r + rowoff) * N + n] = acc[r];
}

// ---------------------------------------------------------------------------
// Embedding gather-means (i1,i2) and l2-normalized means (i3,i4).
// grid (T=128, 4), block 64 (one thread per feature).
// ---------------------------------------------------------------------------
__global__ void k_embed_mean(const int* __restrict__ diag_idx, const int* __restrict__ proc_idx,
                             const float* __restrict__ emb_diag, const float* __restrict__ emb_proc,
                             const float* __restrict__ dm_diag_emb, const float* __restrict__ pm_proc_emb,
                             float* __restrict__ i1, float* __restrict__ i2,
                             float* __restrict__ i3, float* __restrict__ i4) {
  __shared__ float red[64];
  const int t = blockIdx.x, which = blockIdx.y, d = threadIdx.x;
  const int* idx; const float* emb; float* out; bool donorm;
  switch (which) {
    case 0:  idx = diag_idx; emb = emb_diag;    out = i1; donorm = false; break;
    case 1:  idx = proc_idx; emb = emb_proc;    out = i2; donorm = false; break;
    case 2:  idx = diag_idx; emb = dm_diag_emb; out = i3; donorm = true;  break;
    default: idx = proc_idx; emb = pm_proc_emb; out = i4; donorm = true;  break;
  }
  float acc = 0.f;
  for (int l = 0; l < 64; ++l)
    acc += emb[(size_t)idx[t * 64 + l] * 64 + d];
  acc *= (1.f / 64.f);
  if (donorm) {
    red[d] = acc * acc; __syncthreads();
    for (int s = 32; s > 0; s >>= 1) { if (d < s) red[d] += red[d + s]; __syncthreads(); }
    acc /= fmaxf(sqrtf(red[0]), 1e-12f);
  }
  out[t * 64 + d] = acc;
}

// qkv = x @ Wqkv^T + b : (128x64)@(64x192). grid 96, block 256.
__global__ void k_qkv(const float* __restrict__ x, const float* __restrict__ W,
                      const float* __restrict__ b, float* __restrict__ qkv) {
  int i = blockIdx.x * 256 + threadIdx.x;   // t*192 + c
  int t = i / 192, c = i % 192;
  float acc = b[c];
  for (int j = 0; j < 64; ++j) acc += x[t * 64 + j] * W[c * 64 + j];
  qkv[i] = acc;
}

// attention per (t, head). grid (128, 8), block 128 (thread = key position s).
__global__ void k_attn(const float* __restrict__ qkv, float* __restrict__ attn_o) {
  __shared__ float p[128];
  __shared__ float red[128];
  __shared__ float sv;
  const int t = blockIdx.x, h = blockIdx.y, s = threadIdx.x;
  float sc = 0.f;
  for (int dh = 0; dh < 8; ++dh)
    sc += qkv[t * 192 + h * 8 + dh] * qkv[s * 192 + 64 + h * 8 + dh];
  sc *= 0.35355339059327373f;  // 1/sqrt(8)
  red[s] = sc; __syncthreads();
  for (int st = 64; st > 0; st >>= 1) { if (s < st) red[s] = fmaxf(red[s], red[s + st]); __syncthreads(); }
  if (s == 0) sv = red[0]; __syncthreads();
  float e = expf(sc - sv);
  red[s] = e; __syncthreads();
  for (int st = 64; st > 0; st >>= 1) { if (s < st) red[s] += red[s + st]; __syncthreads(); }
  if (s == 0) sv = red[0]; __syncthreads();
  p[s] = e / sv; __syncthreads();
  for (int dh = 0; dh < 8; ++dh) {
    red[s] = p[s] * qkv[s * 192 + 128 + h * 8 + dh];
    __syncthreads();
    for (int st = 64; st > 0; st >>= 1) { if (s < st) red[s] += red[s + st]; __syncthreads(); }
    if (s == 0) attn_o[t * 64 + h * 8 + dh] = red[0];
    __syncthreads();
  }
}

// x2 = LN(x + o @ Wo^T + bo).  grid 128, block 64.
__global__ void k_addln(const float* __restrict__ x, const float* __restrict__ o,
                        const float* __restrict__ Wo, const float* __restrict__ bo,
                        const float* __restrict__ g, const float* __restrict__ b,
                        float* __restrict__ out) {
  __shared__ float red[64];
  __shared__ float smu, svar;
  const int t = blockIdx.x, d = threadIdx.x;
  float acc = bo[d];
  for (int j = 0; j < 64; ++j) acc += o[t * 64 + j] * Wo[d * 64 + j];
  float v = x[t * 64 + d] + acc;
  red[d] = v; __syncthreads();
  for (int s = 32; s > 0; s >>= 1) { if (d < s) red[d] += red[d + s]; __syncthreads(); }
  if (d == 0) smu = red[0] * (1.f / 64.f); __syncthreads();
  float dv = v - smu;
  red[d] = dv * dv; __syncthreads();
  for (int s = 32; s > 0; s >>= 1) { if (d < s) red[d] += red[d + s]; __syncthreads(); }
  if (d == 0) svar = red[0] * (1.f / 64.f); __syncthreads();
  out[t * 64 + d] = dv * rsqrtf(svar + 1e-5f) * g[d] + b[d];
}

// out = LN(x + relu(x@W1^T+b1)@W2^T+b2).  W1:(8,64), W2:(64,8). grid 128, block 64.
__global__ void k_ffln(const float* __restrict__ x,
                       const float* __restrict__ W1, const float* __restrict__ b1,
                       const float* __restrict__ W2, const float* __restrict__ b2,
                       const float* __restrict__ g, const float* __restrict__ b,
                       float* __restrict__ out) {
  __shared__ float h8[8];
  __shared__ float red[64];
  __shared__ float smu, svar;
  const int t = blockIdx.x, d = threadIdx.x;
  if (d < 8) {
    float a = b1[d];
    for (int j = 0; j < 64; ++j) a += x[t * 64 + j] * W1[d * 64 + j];
    h8[d] = fmaxf(a, 0.f);
  }
  __syncthreads();
  float f = b2[d];
  for (int i = 0; i < 8; ++i) f += h8[i] * W2[d * 8 + i];
  float v = x[t * 64 + d] + f;
  red[d] = v; __syncthreads();
  for (int s = 32; s > 0; s >>= 1) { if (d < s) red[d] += red[d + s]; __syncthreads(); }
  if (d == 0) smu = red[0] * (1.f / 64.f); __syncthreads();
  float dv = v - smu;
  red[d] = dv * dv; __syncthreads();
  for (int s = 32; s > 0; s >>= 1) { if (d < s) red[d] += red[d + s]; __syncthreads(); }
  if (d == 0) svar = red[0] * (1.f / 64.f); __syncthreads();
  out[t * 64 + d] = dv * rsqrtf(svar + 1e-5f) * g[d] + b[d];
}

// patient = concat([i3,o1,o2,i4], -1). grid 128, block 256.
__global__ void k_concat_patient(const float* __restrict__ i3, const float* __restrict__ o1,
                                 const float* __restrict__ o2, const float* __restrict__ i4,
                                 float* __restrict__ patient) {
  int t = blockIdx.x, j = threadIdx.x;
  float v;
  if      (j < 64)  v = i3[t * 64 + j];
  else if (j < 128) v = o1[t * 64 + (j - 64)];
  else if (j < 192) v = o2[t * 64 + (j - 128)];
  else              v = i4[t * 64 + (j - 192)];
  patient[t * 256 + j] = v;
}

// head: LN(gelu(LN(in,lnA)) @ W^T + b, lnB). in:(R,256) -> out:(R,64). grid R, block 256.
__global__ void k_head(const float* __restrict__ in, const float* __restrict__ W,
                       const float* __restrict__ bias,
                       const float* __restrict__ lnA_g, const float* __restrict__ lnA_b,
                       const float* __restrict__ lnB_g, const float* __restrict__ lnB_b,
                       float* __restrict__ out) {
  __shared__ float xs[256];
  __shared__ float red[256];
  __shared__ float s0, s1;
  const int t = blockIdx.x, j = threadIdx.x;
  float x = in[t * 256 + j];
  red[j] = x; __syncthreads();
  for (int s = 128; s > 0; s >>= 1) { if (j < s) red[j] += red[j + s]; __syncthreads(); }
  if (j == 0) s0 = red[0] * (1.f / 256.f); __syncthreads();
  float mu = s0, dv = x - mu;
  red[j] = dv * dv; __syncthreads();
  for (int s = 128; s > 0; s >>= 1) { if (j < s) red[j] += red[j + s]; __syncthreads(); }
  if (j == 0) s1 = red[0] * (1.f / 256.f); __syncthreads();
  float yn = dv * rsqrtf(s1 + 1e-5f) * lnA_g[j] + lnA_b[j];
  xs[j] = 0.5f * yn * (1.f + erff(yn * 0.70710678118654752f));  // exact GELU
  __syncthreads();
  float tv = 0.f;
  if (j < 64) {
    tv = bias[j];
    for (int k = 0; k < 256; ++k) tv += xs[k] * W[j * 256 + k];
  }
  red[j] = (j < 64) ? tv : 0.f; __syncthreads();
  for (int s = 32; s > 0; s >>= 1) { if (j < s) red[j] += red[j + s]; __syncthreads(); }
  if (j == 0) s0 = red[0] * (1.f / 64.f); __syncthreads();
  float mu2 = s0, dv2 = tv - mu2;
  red[j] = (j < 64) ? dv2 * dv2 : 0.f; __syncthreads();
  for (int s = 32; s > 0; s >>= 1) { if (j < s) red[j] += red[j + s]; __syncthreads(); }
  if (j == 0) s1 = red[0] * (1.f / 64.f); __syncthreads();
  if (j < 64) out[t * 64 + j] = dv2 * rsqrtf(s1 + 1e-5f) * lnB_g[j] + lnB_b[j];
}

// elementwise epilogues -----------------------------------------------------
__global__ void k_bias_col_relu(float* __restrict__ X, const float* __restrict__ b, int total) {
  int i = blockIdx.x * 256 + threadIdx.x;
  if (i < total) X[i] = fmaxf(X[i] + b[i & 63], 0.f);
}
__global__ void k_combine_dm(const float* __restrict__ P0, const float* __restrict__ P1,
                             const float* __restrict__ b20, const float* __restrict__ b21,
                             const float* __restrict__ inter, float* __restrict__ dm, int total) {
  int i = blockIdx.x * 256 + threadIdx.x;
  if (i < total) {
    int c = i & 63;
    dm[i] = (P0[i] + b20[c]) - inter[0] * (P1[i] + b21[c]);
  }
}
__global__ void k_bias_row512(float* __restrict__ X, const float* __restrict__ b, int total) {
  int i = blockIdx.x * 256 + threadIdx.x;
  if (i < total) X[i] += b[i >> 9];       // row-major (4096,512), bias per row
}
__global__ void k_bias_col64(float* __restrict__ X, const float* __restrict__ b, int total) {
  int i = blockIdx.x * 256 + threadIdx.x;
  if (i < total) X[i] += b[i & 63];
}
__global__ void k_transpose_projW(const float* __restrict__ W, float* __restrict__ Wt) {
  int i = blockIdx.x * 256 + threadIdx.x;  // 512*64
  if (i < 512 * 64) {
    int j = i >> 6, c = i & 63;            // Wt[j][c] = W[c][j], W:(64,512)
    Wt[i] = W[c * 512 + j];
  }
}

// l2-normalize rows of src(4096x64) into slice y of dmcat(4096x256). grid (4096,4), block 64.
__global__ void k_l2ncat(const float* __restrict__ s0, const float* __restrict__ s1,
                         const float* __restrict__ s2, const float* __restrict__ s3,
                         float* __restrict__ dmcat) {
  __shared__ float red[64];
  const int r = blockIdx.x, y = blockIdx.y, d = threadIdx.x;
  const float* src = (y == 0) ? s0 : (y == 1) ? s1 : (y == 2) ? s2 : s3;
  float v = src[(size_t)r * 64 + d];
  red[d] = v * v; __syncthreads();
  for (int s = 32; s > 0; s >>= 1) { if (d < s) red[d] += red[d + s]; __syncthreads(); }
  dmcat[(size_t)r * 256 + y * 64 + d] = v / fmaxf(sqrtf(red[0]), 1e-12f);
}

// fact1 = softmax(query @ dm^T) @ dm over 4096 rows. one block of 1024.
__global__ __launch_bounds__(1024)
void k_fact1(const float* __restrict__ queries, const float* __restrict__ dm,
             float* __restrict__ fact1) {
  __shared__ float sc[4096];
  __shared__ float red[1024];
  __shared__ float sv;
  const int tid = threadIdx.x;
  const float* q = queries + 127 * 64;
  float lmax = -INFINITY;
  for (int n = tid; n < 4096; n += 1024) {
    float s = 0.f;
    for (int c = 0; c < 64; ++c) s += q[c] * dm[(size_t)n * 64 + c];
    sc[n] = s;
    lmax = fmaxf(lmax, s);
  }
  red[tid] = lmax; __syncthreads();
  for (int st = 512; st > 0; st >>= 1) { if (tid < st) red[tid] = fmaxf(red[tid], red[tid + st]); __syncthreads(); }
  if (tid == 0) sv = red[0]; __syncthreads();
  float mx = sv, lsum = 0.f;
  for (int n = tid; n < 4096; n += 1024) {
    float e = expf(sc[n] - mx);
    sc[n] = e;
    lsum += e;
  }
  red[tid] = lsum; __syncthreads();
  for (int st = 512; st > 0; st >>= 1) { if (tid < st) red[tid] += red[tid + st]; __syncthreads(); }
  if (tid == 0) sv = red[0]; __syncthreads();
  float inv = 1.f / sv;
  const int c = tid & 63, part = tid >> 6;   // 16 partials per column
  float acc = 0.f;
  for (int n = part; n < 4096; n += 16) acc += sc[n] * dm[(size_t)n * 64 + c];
  red[part * 64 + c] = acc; __syncthreads();
  for (int st = 8; st > 0; st >>= 1) { if (part < st) red[part * 64 + c] += red[(part + st) * 64 + c]; __syncthreads(); }
  if (part == 0) fact1[c] = red[c] * inv;
}

// rowsum[t] = (onehot(med_hist_idx[t]) @ dm), with .set() dedupe. grid 127, block 64.
__global__ void k_rowsum(const int* __restrict__ med_hist, const float* __restrict__ dm,
                         float* __restrict__ rowsum) {
  const int t = blockIdx.x, c = threadIdx.x;
  float acc = 0.f;
  for (int l = 0; l < 48; ++l) {
    int idx = med_hist[t * 48 + l];
    bool dup = false;
    for (int l2 = 0; l2 < l; ++l2) dup |= (med_hist[t * 48 + l2] == idx);
    if (!dup) acc += dm[(size_t)idx * 64 + c];
  }
  rowsum[t * 64 + c] = acc;
}

// fact2 = softmax(query @ hk^T) @ rowsum. one block of 128.
__global__ void k_fact2(const float* __restrict__ queries, const float* __restrict__ rowsum,
                        float* __restrict__ fact2) {
  __shared__ float p[128];
  __shared__ float red[128];
  __shared__ float sv;
  const int t = threadIdx.x;
  const float* q = queries + 127 * 64;
  float s = -INFINITY;
  if (t < 127) {
    s = 0.f;
    for (int c = 0; c < 64; ++c) s += q[c] * queries[t * 64 + c];
  }
  red[t] = s; __syncthreads();
  for (int st = 64; st > 0; st >>= 1) { if (t < st) red[t] = fmaxf(red[t], red[t + st]); __syncthreads(); }
  if (t == 0) sv = red[0]; __syncthreads();
  float e = (t < 127) ? expf(s - sv) : 0.f;
  red[t] = e; __syncthreads();
  for (int st = 64; st > 0; st >>= 1) { if (t < st) red[t] += red[t + st]; __syncthreads(); }
  if (t == 0) sv = red[0]; __syncthreads();
  p[t] = e / sv; __syncthreads();
  if (t < 64) {
    float acc = 0.f;
    for (int tt = 0; tt < 127; ++tt) acc += p[tt] * rowsum[tt * 64 + t];
    fact2[t] = acc;
  }
}

// h = relu(relu(concat) @ out_W1^T + out_b1). one block of 128.
__global__ void k_out_hidden(const float* __restrict__ queries, const float* __restrict__ fact1,
                             const float* __restrict__ fact2, const float* __restrict__ W1,
                             const float* __restrict__ b1, float* __restrict__ h) {
  const int i = threadIdx.x;
  const float* q = queries + 127 * 64;
  float acc = b1[i];
  for (int j = 0; j < 192; ++j) {
    float xv = (j < 64) ? q[j] : (j < 128) ? fact1[j - 64] : fact2[j - 128];
    acc += fmaxf(xv, 0.f) * W1[i * 192 + j];
  }
  h[i] = fmaxf(acc, 0.f);
}

// out = h @ out_W2^T + out_b2.  grid 16, block 256.
__global__ void k_out_final(const float* __restrict__ h, const float* __restrict__ W2,
                            const float* __restrict__ b2, float* __restrict__ out) {
  const int n = blockIdx.x * 256 + threadIdx.x;
  float acc = b2[n];
  for (int j = 0; j < 128; ++j) acc += h[j] * W2[(size_t)n * 128 + j];
  out[n] = acc;
}

// ---------------------------------------------------------------------------
extern "C" void kernel_launch(void* const* d_in, const int* in_sizes, int n_in,
                              void* d_out, int out_size, void* d_ws, size_t ws_size,
                              hipStream_t stream) {
  const int*   diag_idx     = (const int*)d_in[0];
  const int*   proc_idx     = (const int*)d_in[1];
  const int*   med_hist     = (const int*)d_in[2];
  const float* emb_diag     = (const float*)d_in[3];
  const float* emb_proc     = (const float*)d_in[4];
  const float* dm_diag_emb  = (const float*)d_in[5];
  const float* pm_proc_emb  = (const float*)d_in[6];
  const float* dm_med_emb   = (const float*)d_in[7];
  const float* pm_med_emb   = (const float*)d_in[8];
  const float* molecule     = (const float*)d_in[9];
  const float* proj_W       = (const float*)d_in[10];
  const float* proj_b       = (const float*)d_in[11];
  const float* projection_W = (const float*)d_in[12];
  const float* projection_b = (const float*)d_in[13];
  const float* gcn_W1       = (const float*)d_in[14];
  const float* gcn_b1       = (const float*)d_in[15];
  const float* gcn_W2       = (const float*)d_in[16];
  const float* gcn_b2       = (const float*)d_in[17];
  const float* enc_Wqkv     = (const float*)d_in[18];
  const float* enc_bqkv     = (const float*)d_in[19];
  const float* enc_Wo       = (const float*)d_in[20];
  const float* enc_bo       = (const float*)d_in[21];
  const float* enc_ff_W1    = (const float*)d_in[22];
  const float* enc_ff_b1    = (const float*)d_in[23];
  const float* enc_ff_W2    = (const float*)d_in[24];
  const float* enc_ff_b2    = (const float*)d_in[25];
  const float* q_W          = (const float*)d_in[26];
  const float* q_b          = (const float*)d_in[27];
  const float* f_W          = (const float*)d_in[28];
  const float* f_b          = (const float*)d_in[29];
  const float* out_W1       = (const float*)d_in[30];
  const float* out_b1       = (const float*)d_in[31];
  const float* out_W2       = (const float*)d_in[32];
  const float* out_b2       = (const float*)d_in[33];
  const float* enc_ln1      = (const float*)d_in[34];
  const float* enc_ln2      = (const float*)d_in[35];
  const float* q_ln1        = (const float*)d_in[36];
  const float* q_ln2        = (const float*)d_in[37];
  const float* f_ln1        = (const float*)d_in[38];
  const float* f_ln2        = (const float*)d_in[39];
  const float* adj_norm     = (const float*)d_in[40];
  const float* inter        = (const float*)d_in[41];

  // workspace carve (floats)
  float* w = (float*)d_ws;
  size_t off = 0;
  auto alloc = [&](size_t n) { float* p = w + off; off += n; return p; };
  float* i1      = alloc(128 * 64);
  float* i2      = alloc(128 * 64);
  float* i3      = alloc(128 * 64);
  float* i4      = alloc(128 * 64);
  float* qkv     = alloc(128 * 192);
  float* attn_o  = alloc(128 * 64);
  float* xa      = alloc(128 * 64);
  float* o1      = alloc(128 * 64);
  float* o2      = alloc(128 * 64);
  float* patient = alloc(128 * 256);
  float* queries = alloc(128 * 64);
  float* Hb      = alloc(2 * 4096 * 64);
  float* Gb      = alloc(2 * 4096 * 64);
  float* Pb      = alloc(2 * 4096 * 64);
  float* dm_raw  = alloc(4096 * 64);
  float* PT      = alloc(4096 * 512);       // (proj_W@molecule)+proj_b, row-major
  float* dmcat   = PT;                      // alias: PT dead after `mol` GEMM
  float* projWt  = alloc(512 * 64);
  float* mol     = alloc(4096 * 64);
  float* dm_fin  = alloc(4096 * 64);
  float* fact1   = alloc(64);
  float* rowsum  = alloc(127 * 64);
  float* fact2   = alloc(64);
  float* hbuf    = alloc(128);
  (void)ws_size; (void)in_sizes; (void)n_in; (void)out_size;

  // 1) embedding means (+ l2n for i3/i4)
  k_embed_mean<<<dim3(128, 4), 64, 0, stream>>>(diag_idx, proc_idx, emb_diag, emb_proc,
                                                dm_diag_emb, pm_proc_emb, i1, i2, i3, i4);

  // 2) two encoder layers
  for (int g = 0; g < 2; ++g) {
    const float* x    = g ? i2 : i1;
    float*       outp = g ? o2 : o1;
    k_qkv<<<96, 256, 0, stream>>>(x, enc_Wqkv + (size_t)g * 192 * 64, enc_bqkv + g * 192, qkv);
    k_attn<<<dim3(128, 8), 128, 0, stream>>>(qkv, attn_o);
    k_addln<<<128, 64, 0, stream>>>(x, attn_o, enc_Wo + (size_t)g * 64 * 64, enc_bo + g * 64,
                                    enc_ln1 + g * 128, enc_ln1 + g * 128 + 64, xa);
    k_ffln<<<128, 64, 0, stream>>>(xa, enc_ff_W1 + (size_t)g * 8 * 64, enc_ff_b1 + g * 8,
                                   enc_ff_W2 + (size_t)g * 64 * 8, enc_ff_b2 + g * 64,
                                   enc_ln2 + g * 128, enc_ln2 + g * 128 + 64, outp);
  }

  // 3) patient + queries head
  k_concat_patient<<<128, 256, 0, stream>>>(i3, o1, o2, i4, patient);
  k_head<<<128, 256, 0, stream>>>(patient, q_W, q_b, q_ln1, q_ln1 + 256, q_ln2, q_ln2 + 64, queries);

  // 4) GCNs: big WMMA GEMMs against the 4096x4096 adjacency matrices
  for (int g = 0; g < 2; ++g) {
    const float* adj = adj_norm + (size_t)g * 4096 * 4096;
    float* H = Hb + (size_t)g * 4096 * 64;
    float* G = Gb + (size_t)g * 4096 * 64;
    float* P = Pb + (size_t)g * 4096 * 64;
    gemm_f32_wmma<<<dim3(256, 1), 128, 0, stream>>>(adj, gcn_W1 + (size_t)g * 4096 * 64, H, 4096, 64, 4096);
    k_bias_col_relu<<<(4096 * 64) / 256, 256, 0, stream>>>(H, gcn_b1 + g * 64, 4096 * 64);
    gemm_f32_wmma<<<dim3(256, 1), 128, 0, stream>>>(H, gcn_W2 + (size_t)g * 64 * 64, G, 4096, 64, 64);
    gemm_f32_wmma<<<dim3(256, 1), 128, 0, stream>>>(adj, G, P, 4096, 64, 4096);
  }
  k_combine_dm<<<(4096 * 64) / 256, 256, 0, stream>>>(Pb, Pb + (size_t)4096 * 64,
                                                      gcn_b2, gcn_b2 + 64, inter, dm_raw, 4096 * 64);

  // 5) molecule path: PT = proj_W @ molecule (+proj_b/row); mol = PT @ projection_W^T + projection_b
  gemm_f32_wmma<<<dim3(256, 8), 128, 0, stream>>>(proj_W, molecule, PT, 4096, 512, 4096);
  k_bias_row512<<<(4096 * 512) / 256, 256, 0, stream>>>(PT, proj_b, 4096 * 512);
  k_transpose_projW<<<(512 * 64) / 256, 256, 0, stream>>>(projection_W, projWt);
  gemm_f32_wmma<<<dim3(256, 1), 128, 0, stream>>>(PT, projWt, mol, 4096, 64, 512);
  k_bias_col64<<<(4096 * 64) / 256, 256, 0, stream>>>(mol, projection_b, 4096 * 64);

  // 6) dm head over l2n-concat (dmcat aliases PT, which is now dead)
  k_l2ncat<<<dim3(4096, 4), 64, 0, stream>>>(dm_med_emb, dm_raw, mol, pm_med_emb, dmcat);
  k_head<<<4096, 256, 0, stream>>>(dmcat, f_W, f_b, f_ln1, f_ln1 + 256, f_ln2, f_ln2 + 64, dm_fin);

  // 7) facts + output MLP
  k_fact1<<<1, 1024, 0, stream>>>(queries, dm_fin, fact1);
  k_rowsum<<<127, 64, 0, stream>>>(med_hist, dm_fin, rowsum);
  k_fact2<<<1, 128, 0, stream>>>(queries, rowsum, fact2);
  k_out_hidden<<<1, 128, 0, stream>>>(queries, fact1, fact2, out_W1, out_b1, hbuf);
  k_out_final<<<16, 256, 0, stream>>>(hbuf, out_W2, out_b2, (float*)d_out);
}